// SOM_9706626090014
// MI455X (gfx1250) — compile-verified
//
#include <hip/hip_runtime.h>
#include <stdint.h>

// ---------------------------------------------------------------------------
// SOM step, MI455X (gfx1250, wave32).
//  Pass 1: score[n] = ||W_n||^2 - 2*(W x)[n]  (argmin-equivalent to L2 dist).
//          W·x via V_WMMA_F32_16X16X4_F32 (dual accumulator chains),
//          ||W||^2 via VALU on the same LDS-staged data.
//          K chunked into 4x128-float tiles; staging via the Tensor Data
//          Mover (TENSORcnt-paced, double-buffered, D# pad -> 132-float
//          pitch) using the probe-verified builtin arity for each toolchain
//          (5-arg on clang-22, 6-arg on clang-23+); else direct load->LDS.
//          Hierarchical packed-u64 argmin: wave shuffle -> 64 buckets -> 1.
//  Pass 2: out = W + lr(n)*(x - W), float4 streaming; W re-read hits L2
//          (128 MB < 192 MB global L2), so HBM traffic ~= 256 MB total.
// ---------------------------------------------------------------------------

typedef float v2f __attribute__((ext_vector_type(2)));
typedef float v8f __attribute__((ext_vector_type(8)));

#define DIM        512
#define NNEUR      65536
#define WAVES_PB   2
#define ROWS_PW    16
#define ROWS_PB    (WAVES_PB * ROWS_PW)   // 32 rows per block
#define CHUNK      128                    // K floats per staged chunk
#define NCHUNK     (DIM / CHUNK)          // 4
#define CPITCH     132                    // 128 + 4 pad: bank-conflict-free
#define BUF_F      (ROWS_PW * CPITCH)     // 2112 floats per chunk buffer
#define WAVE_F     (2 * BUF_F + DIM)      // 4736 floats per wave
#define SMEM_F     (WAVES_PB * WAVE_F)    // ~37.9 KB per block
#define NBUCKETS   64

#if __has_builtin(__builtin_amdgcn_wmma_f32_16x16x4_f32)
#define HAVE_WMMA_F32X4 1
#else
#define HAVE_WMMA_F32X4 0
#endif

#if __has_builtin(__builtin_amdgcn_tensor_load_to_lds) && \
    __has_builtin(__builtin_amdgcn_s_wait_tensorcnt)
#define USE_TDM 1
typedef unsigned int u32x4 __attribute__((ext_vector_type(4)));
typedef int          i32x4 __attribute__((ext_vector_type(4)));
typedef int          i32x8 __attribute__((ext_vector_type(8)));

// D# for a 16-row x 128-col f32 tile of a row-major [*,512] tensor, with
// 4-dword LDS padding after every 512-byte row (-> pitch 132 floats).
__device__ __forceinline__ void tdm_load_tile16x128(const float* gsrc,
                                                    unsigned lds_byte) {
    unsigned long long ga = (unsigned long long)(uintptr_t)gsrc;
    u32x4 g0;
    g0.x = 1u;                                   // count=1 (valid), no gather
    g0.y = lds_byte;                             // lds_addr [63:32]
    g0.z = (unsigned)ga;                         // global_addr[31:0]
    g0.w = (unsigned)((ga >> 32) & 0x01FFFFFFu)  // global_addr[56:32]
         | (2u << 30);                           // type=2 ("image")
    i32x8 g1;
    g1[0] = (2 << 16)        // data_size = 4 B
          | (1 << 20)        // pad_enable
          | (6 << 22)        // pad_interval: 2^6 qwords = 512 B (one row)
          | (3 << 25);       // pad_amount: 4 dwords
    g1[1] = (DIM & 0xFFFF) << 16;        // tensor_dim0[15:0] = 512
    g1[2] = (ROWS_PW & 0xFFFF) << 16;    // tensor_dim0[31:16]=0 | tensor_dim1=16
    g1[3] = (CHUNK & 0xFFFF) << 16;      // tensor_dim1[31:16]=0 | tile_dim0=128
    g1[4] = ROWS_PW;                     // tile_dim1=16 | tile_dim2=0
    g1[5] = DIM;                         // tensor_dim0_stride[31:0] = 512
    g1[6] = 0;                           // stride0[47:32] | stride1[15:0]
    g1[7] = 0;                           // stride1[47:16]
    i32x4 z4 = {0, 0, 0, 0};             // groups 2/3 unused (2D tile)
#if __clang_major__ >= 23
    i32x8 z8 = {0, 0, 0, 0, 0, 0, 0, 0};
    __builtin_amdgcn_tensor_load_to_lds(g0, g1, z4, z4, z8, 0);
#else
    __builtin_amdgcn_tensor_load_to_lds(g0, g1, z4, z4, 0);
#endif
}
#else
#define USE_TDM 0
#endif

__global__ void som_init(unsigned long long* ws) {
    int t = threadIdx.x;
    if (t <= NBUCKETS) ws[t] = ~0ULL;
}

__global__ void __launch_bounds__(WAVES_PB * 32)
som_pass1(const float* __restrict__ x,
          const float* __restrict__ W,
          unsigned long long* __restrict__ buckets) {
    extern __shared__ float smem[];
    const int lane    = threadIdx.x & 31;
    const int wave    = threadIdx.x >> 5;
    const int rowBase = blockIdx.x * ROWS_PB + wave * ROWS_PW;

    float* buf0 = smem + wave * WAVE_F;
    float* buf1 = buf0 + BUF_F;
    float* xs   = buf0 + 2 * BUF_F;

    // ---- stage x into this wave's LDS region (barrier-free, per-wave) ----
#pragma unroll
    for (int i = 0; i < DIM / 128; ++i)
        ((float4*)xs)[i * 32 + lane] = ((const float4*)x)[i * 32 + lane];

    const float* gbase = W + (size_t)rowBase * DIM;
    const int koff = (lane >= 16) ? 2 : 0;     // A/B K-offset for upper half-wave
    const int arow = lane & 15;

    v8f   acc0 = {}, acc1 = {};                // two WMMA chains (halve RAW dep)
    float nrm  = 0.f;
    float dotp = 0.f;

    // Consume one staged 16x128 chunk: 32 WMMA (16x16x4 f32) + norm FMAs.
    auto consume = [&](const float* buf, int kbase) {
        const float* ap = buf + arow * CPITCH;
        const float* bp = xs + kbase;
#pragma unroll
        for (int k0 = 0; k0 < CHUNK; k0 += 8) {
            int kk = k0 + koff;
            v2f a0 = *(const v2f*)(ap + kk);
            v2f b0 = *(const v2f*)(bp + kk);
            v2f a1 = *(const v2f*)(ap + kk + 4);
            v2f b1 = *(const v2f*)(bp + kk + 4);
            nrm += a0.x * a0.x + a0.y * a0.y + a1.x * a1.x + a1.y * a1.y;
#if HAVE_WMMA_F32X4
            acc0 = __builtin_amdgcn_wmma_f32_16x16x4_f32(
                       false, a0, false, b0, (short)0, acc0, false, false);
            acc1 = __builtin_amdgcn_wmma_f32_16x16x4_f32(
                       false, a1, false, b1, (short)0, acc1, false, false);
#else
            dotp += a0.x * b0.x + a0.y * b0.y + a1.x * b1.x + a1.y * b1.y;
#endif
        }
    };

#if USE_TDM
    // ---- Tensor Data Mover pipeline: double-buffered, TENSORcnt-paced ----
    // lds_addr from pointer low bits (workgroup-relative LDS byte offset).
    unsigned off0 = (unsigned)(uintptr_t)buf0;
    unsigned off1 = (unsigned)(uintptr_t)buf1;
    tdm_load_tile16x128(gbase, off0);                          // chunk 0
#pragma unroll 1
    for (int c = 0; c < NCHUNK - 1; ++c) {
        tdm_load_tile16x128(gbase + (c + 1) * CHUNK,
                            ((c + 1) & 1) ? off1 : off0);
        __builtin_amdgcn_s_wait_tensorcnt(1);  // chunk c landed, c+1 in flight
        asm volatile("" ::: "memory");         // keep LDS reads below the wait
        consume((c & 1) ? buf1 : buf0, c * CHUNK);
    }
    __builtin_amdgcn_s_wait_tensorcnt(0);      // last chunk landed
    asm volatile("" ::: "memory");
    consume(((NCHUNK - 1) & 1) ? buf1 : buf0, (NCHUNK - 1) * CHUNK);
#else
    // ---- direct staging: coalesced global b128 -> LDS, chunk at a time ----
    // Rolled outer loop keeps one chunk live (no spills); DS in-order makes
    // the single per-wave buffer race-free; other waves hide the latency.
#pragma unroll 1
    for (int c = 0; c < NCHUNK; ++c) {
        const float4* g4 = (const float4*)(gbase + c * CHUNK);
#pragma unroll 4
        for (int i = 0; i < 16; ++i) {
            int idx = i * 32 + lane;
            float4 v = g4[(idx >> 5) * (DIM / 4) + (idx & 31)];
            *(float4*)(buf0 + (idx >> 5) * CPITCH + (idx & 31) * 4) = v;
        }
        consume(buf0, c * CHUNK);
    }
#endif

    // full row norm: lane m and m+16 each hold half of row (m&15)
    float nrm2 = nrm + __shfl_xor(nrm, 16);

#if HAVE_WMMA_F32X4
    v8f acc = acc0 + acc1;
    // C/D layout: vgpr r, lanes0-15 -> M=r; lanes16-31 -> M=r+8 (all N equal).
    bool active = (lane & 8) == 0;                 // lanes 0-7, 16-23
    int  m      = (lane & 7) | ((lane & 16) >> 1); // 0-7 or 8-15
    float dot = 0.f;
#pragma unroll
    for (int i = 0; i < 8; ++i)
        if ((lane & 7) == i) dot = acc[i];
    (void)dotp;
#else
    bool active = lane < 16;
    int  m      = lane;
    float dot   = dotp + __shfl_xor(dotp, 16);
    (void)acc0; (void)acc1;
#endif
    float normv = __shfl(nrm2, m & 15);

    float score = active ? (normv - 2.f * dot) : __builtin_inff();
    int   rown  = active ? (rowBase + m)       : 0x7FFFFFFF;

    // wave (score, index) min-reduce; tie -> lower index (matches argmin)
#pragma unroll
    for (int off = 16; off; off >>= 1) {
        float os  = __shfl_xor(score, off);
        int   orn = __shfl_xor(rown, off);
        if (os < score || (os == score && orn < rown)) { score = os; rown = orn; }
    }

    if (lane == 0) {
        unsigned u = __float_as_uint(score);
        unsigned mask = (unsigned)(((int)u) >> 31) | 0x80000000u; // order-preserving
        u ^= mask;
        unsigned long long key =
            ((unsigned long long)u << 32) | (unsigned)rown;
        atomicMin(&buckets[blockIdx.x & (NBUCKETS - 1)], key);
    }
}

__global__ void som_reduce(unsigned long long* ws) {
    int lane = threadIdx.x;                       // 32 threads
    unsigned long long a = ws[lane];
    unsigned long long b = ws[lane + 32];
    unsigned long long k = (b < a) ? b : a;
#pragma unroll
    for (int off = 16; off; off >>= 1) {
        unsigned lo = (unsigned)k, hi = (unsigned)(k >> 32);
        lo = __shfl_xor(lo, off);
        hi = __shfl_xor(hi, off);
        unsigned long long o = ((unsigned long long)hi << 32) | lo;
        if (o < k) k = o;
    }
    if (lane == 0) ws[NBUCKETS] = k;
}

__global__ void __launch_bounds__(128)
som_pass2(const float* __restrict__ x,
          const float* __restrict__ W,
          const int* __restrict__ pos,
          const int* __restrict__ itp,
          const unsigned long long* __restrict__ ws,
          float* __restrict__ out) {
    const int n = blockIdx.x;
    const int t = threadIdx.x;          // 128 threads, float4 each -> 512

    unsigned bmu = (unsigned)(ws[NBUCKETS] & 0xFFFFFFFFu);
    float bi = (float)pos[2 * bmu], bj = (float)pos[2 * bmu + 1];
    float pi = (float)pos[2 * n],   pj = (float)pos[2 * n + 1];
    float di = pi - bi, dj = pj - bj;
    float d2 = di * di + dj * dj;

    float decay = 1.0f - (float)itp[0] * (1.0f / 1000.0f);
    float alpha = 0.3f * decay;
    float sigma = 128.0f * decay;
    float lr    = alpha * __expf(-d2 / (sigma * sigma));

    size_t base = (size_t)n * DIM;
    float4 wv = ((const float4*)(W + base))[t];   // L2-resident after pass 1
    float4 xv = ((const float4*)x)[t];
    float4 o;
    o.x = fmaf(lr, xv.x - wv.x, wv.x);
    o.y = fmaf(lr, xv.y - wv.y, wv.y);
    o.z = fmaf(lr, xv.z - wv.z, wv.z);
    o.w = fmaf(lr, xv.w - wv.w, wv.w);
    ((float4*)(out + base))[t] = o;
}

extern "C" void kernel_launch(void* const* d_in, const int* in_sizes, int n_in,
                              void* d_out, int out_size, void* d_ws, size_t ws_size,
                              hipStream_t stream) {
    (void)in_sizes; (void)n_in; (void)out_size; (void)ws_size;
    const float* x   = (const float*)d_in[0];
    const float* W   = (const float*)d_in[1];
    const int*   pos = (const int*)d_in[2];
    const int*   it  = (const int*)d_in[3];
    unsigned long long* ws = (unsigned long long*)d_ws;
    float* out = (float*)d_out;

    som_init<<<1, 128, 0, stream>>>(ws);
    som_pass1<<<NNEUR / ROWS_PB, WAVES_PB * 32,
                SMEM_F * sizeof(float), stream>>>(x, W, ws);
    som_reduce<<<1, 32, 0, stream>>>(ws);
    som_pass2<<<NNEUR, 128, 0, stream>>>(x, W, pos, it, ws, out);
}